// ExponentialMovingAverage_70257075028064
// MI455X (gfx1250) — compile-verified
//
#include <hip/hip_runtime.h>
#include <hip/hip_bf16.h>
#include <math.h>

// ---------------------------------------------------------------------------
// EMA scan with bias correction:  out[r,t] = corr[t] * sum_{k<=t} m*(1-m)^(t-k) x[r,k]
// corr[t] = 1/(1-(1-m)^(t+1)), m = 0.01.
//
// MI455X (gfx1250): memory bound, 512 MiB traffic -> ~22us floor @ 23.3 TB/s.
//  - per wave: 16 independent rows; time chunked by 16; chunk-local triangular
//    scan = 4 chained V_WMMA_F32_16X16X4_F32 (rows as M, time as N).
//  - inter-chunk carry: y = D + (1-m)^(n+1)*s ; s re-broadcast from lane 15/31
//    via v_readlane + cndmask (bpermute version was mis-merged by the optimizer).
//  - global->LDS staging with async b128 loads, double buffered,
//    s_wait_asynccnt; LDS rows padded to 68 floats (16B-aligned b128 writes,
//    bank-conflict-free b64 reads). Wave-private LDS -> zero barriers.
// ---------------------------------------------------------------------------

typedef __attribute__((ext_vector_type(2))) float v2f;
typedef __attribute__((ext_vector_type(4))) float v4f;
typedef __attribute__((ext_vector_type(8))) float v8f;

#define MOM 0.01f
#define OM  0.99f
#define OM16 0.85145777f   /* 0.99^16 */
#define T_LEN 8192
#define TILE_T 64          /* timesteps staged per LDS tile   */
#define CHUNK 16           /* timesteps per WMMA chain        */
#define RS 68              /* LDS row stride (floats) */
#define ROWS_PER_WAVE 16
#define WAVES_PER_BLOCK 4

#if __has_builtin(__builtin_amdgcn_global_load_async_to_lds_b128)
#define HAS_ASYNC 1
typedef int v4i_vs __attribute__((vector_size(16)));
typedef __attribute__((address_space(1))) v4i_vs gvec4;
typedef __attribute__((address_space(3))) v4i_vs lvec4;
#else
#define HAS_ASYNC 0
#endif

#if __has_builtin(__builtin_amdgcn_s_wait_asynccnt)
#define WAIT_ASYNC(nimm) do { __builtin_amdgcn_s_wait_asynccnt(nimm); asm volatile("" ::: "memory"); } while (0)
#elif HAS_ASYNC
#define WAIT_ASYNC(nimm) asm volatile("s_wait_asynccnt %0" :: "i"(nimm) : "memory")
#else
#define WAIT_ASYNC(nimm) do { } while (0)
#endif

__global__ __launch_bounds__(WAVES_PER_BLOCK * 32)
void ema_scan_wmma_kernel(const float* __restrict__ x, float* __restrict__ out, int rows) {
    __shared__ float sm[WAVES_PER_BLOCK * 2 * ROWS_PER_WAVE * RS];

    const int lane  = threadIdx.x & 31;
    const int wslot = threadIdx.x >> 5;
    const int wid   = blockIdx.x * WAVES_PER_BLOCK + wslot;
    const int r0    = wid * ROWS_PER_WAVE;
    if (r0 >= rows) return;                 // uniform per wave -> EXEC stays all-ones

    float* smw = &sm[wslot * (2 * ROWS_PER_WAVE * RS)];
    const int half = lane >> 4;             // 0: lanes 0-15, 1: lanes 16-31
    const int n    = lane & 15;             // N index (time within chunk) / A row

    // ---- constant B = T^T slices for the 4 chained K=4 WMMAs -------------
    // B (4x16 f32): VGPR j, lanes 0-15 -> K = 2*half + j.
    // B[q][j] = T^T[k, n] = (n >= k) ? m*(1-m)^(n-k) : 0,  k = 4q + 2*half + j.
    v2f B[4];
    #pragma unroll
    for (int q = 0; q < 4; ++q) {
        #pragma unroll
        for (int j = 0; j < 2; ++j) {
            const int k = 4 * q + 2 * half + j;
            B[q][j] = (n >= k) ? MOM * powf(OM, (float)(n - k)) : 0.0f;
        }
    }
    const float dlane = powf(OM, (float)(n + 1));   // (1-m)^(n+1)

    // carry s[v]: EMA value at end of previous chunk, C/D layout
    // (VGPR v, lanes 0-15 -> row v, lanes 16-31 -> row v+8).
    v8f s = {0.f, 0.f, 0.f, 0.f, 0.f, 0.f, 0.f, 0.f};
    float p0 = 1.0f;                                 // (1-m)^(chunk start t)

    // ---- async tile staging: 16 rows x 64 floats -> LDS ------------------
    auto issue = [&](int tile, int buf) {
        const int t0 = tile * TILE_T;
        float* dst = smw + buf * (ROWS_PER_WAVE * RS);
        #pragma unroll
        for (int p = 0; p < 8; ++p) {
            const int row = 2 * p + half;            // 2 rows per instruction
            const float* g = x + (size_t)(r0 + row) * T_LEN + t0 + n * 4;
            float*       l = dst + row * RS + n * 4;
#if HAS_ASYNC
            __builtin_amdgcn_global_load_async_to_lds_b128((gvec4*)g, (lvec4*)l, 0, 0);
#else
            *(v4f*)l = *(const v4f*)g;               // synchronous fallback
#endif
        }
    };

    const int NT = T_LEN / TILE_T;
    issue(0, 0);
    for (int tile = 0; tile < NT; ++tile) {
        const int  cur  = tile & 1;
        const bool more = (tile + 1) < NT;
        if (more) issue(tile + 1, cur ^ 1);          // prefetch next tile
        if (more) { WAIT_ASYNC(8); } else { WAIT_ASYNC(0); }

        const float* src = smw + cur * (ROWS_PER_WAVE * RS);

        #pragma unroll
        for (int sub = 0; sub < 4; ++sub) {
            const int tb = sub * CHUNK;

            // A (16x4 f32): lane m = n (row), VGPR j -> K = 2*half + j
            v8f acc = {0.f, 0.f, 0.f, 0.f, 0.f, 0.f, 0.f, 0.f};
            #pragma unroll
            for (int q = 0; q < 4; ++q) {
                const v2f a = *(const v2f*)(src + n * RS + tb + 4 * q + 2 * half);
                acc = __builtin_amdgcn_wmma_f32_16x16x4_f32(
                        false, a, false, B[q], (short)0, acc, false, false);
            }

            // bias correction: corr = 1/(1 - (1-m)^(tg+n+1)) = 1/(1 - p0*dlane)
#if __has_builtin(__builtin_amdgcn_rcpf)
            const float rc = __builtin_amdgcn_rcpf(1.0f - p0 * dlane);
#else
            const float rc = 1.0f / (1.0f - p0 * dlane);
#endif
            const int tg = tile * TILE_T + tb;

            v8f y;
            #pragma unroll
            for (int v = 0; v < 8; ++v) {
                y[v] = acc[v] + dlane * s[v];         // add decayed carry
                out[(size_t)(r0 + v + 8 * half) * T_LEN + tg + n] = y[v] * rc;
            }

            // new carry: y at last timestep of chunk (lane 15 -> rows 0-7,
            // lane 31 -> rows 8-15). Use readlane + per-half select; the empty
            // asm pins each y[v] as a distinct value so the 8 broadcasts can't
            // be erroneously CSE'd into one (as happened with ds_bpermute).
            #pragma unroll
            for (int v = 0; v < 8; ++v) {
                float t = y[v];
                asm("" : "+v"(t));
                const int lo = __builtin_amdgcn_readlane(__builtin_bit_cast(int, t), 15);
                const int hi = __builtin_amdgcn_readlane(__builtin_bit_cast(int, t), 31);
                s[v] = __builtin_bit_cast(float, half ? hi : lo);
            }
            p0 *= OM16;
        }
    }
}

extern "C" void kernel_launch(void* const* d_in, const int* in_sizes, int n_in,
                              void* d_out, int out_size, void* d_ws, size_t ws_size,
                              hipStream_t stream) {
    (void)n_in; (void)out_size; (void)d_ws; (void)ws_size;
    const float* x  = (const float*)d_in[0];
    float*       out = (float*)d_out;
    const int rows  = in_sizes[0] / T_LEN;            // 32*256 = 8192 rows of length 8192
    const int waves = (rows + ROWS_PER_WAVE - 1) / ROWS_PER_WAVE;
    const int blocks = (waves + WAVES_PER_BLOCK - 1) / WAVES_PER_BLOCK;
    ema_scan_wmma_kernel<<<dim3(blocks), dim3(WAVES_PER_BLOCK * 32), 0, stream>>>(x, out, rows);
}